// Entangle_layer_16140487098687
// MI455X (gfx1250) — compile-verified
//
#include <hip/hip_runtime.h>

// ---------------------------------------------------------------------------
// Entangle layer, 14 qubits, 4 blocks, dc 1->16.  Single fused launch.
// out[b,k,d,x] = (product of <=2 commuting 2x2 local unitaries) applied to
// in[b,k,0,:], where the unitaries depend on only 2 bits of d per block
// (the other 2 d-bits are pure replication -> compute 4 variants, write x4).
//   cI = diag(1,-i), cZ = diag(1,+i)            (diagonal -> scale by {1,+-i})
//   tI = 0.5[[1-i,1+i],[1+i,1-i]], tZ = conj(tI) (= tI with outputs swapped)
// x-bit for qubit q is bit (13-q).
//   block0: scale@bit13 (d-bit0), scale@bit6  (d-bit2)   free d-bits 1,3
//   block1: dense@bit0  (d-bit0), scale@bit11 (d-bit1)   free d-bits 2,3
//   block2: dense@bit8  (d-bit1), scale@bit3  (d-bit3)   free d-bits 0,2
//   block3: dense@bit12 (d-bit2), dense@bit1  (d-bit3)   free d-bits 0,1
// Roofline: 8 MiB read + 128 MiB NT write -> ~6 us at 23.3 TB/s. AI ~ 1-3
// flop/byte << f32 machine balance, so WMMA is pointless; the CDNA5 feature
// that matters is the async-to-LDS DMA path (block2) + b128 NT streams.
// ---------------------------------------------------------------------------

typedef __attribute__((ext_vector_type(4))) float f4;   // two complex64

#define DIM4 8192u   // 16384 complex / 2 per f4

// a*u + b*v per complex lane, a=(1-i)/2, b=(1+i)/2  (row of tI)
__device__ __forceinline__ f4 cmix(f4 u, f4 v) {
  f4 r;
  r.x = 0.5f * (u.x + u.y + v.x - v.y);
  r.y = 0.5f * (u.y - u.x + v.x + v.y);
  r.z = 0.5f * (u.z + u.w + v.z - v.w);
  r.w = 0.5f * (u.w - u.z + v.z + v.w);
  return r;
}

// multiply both complex lanes by (fr + i*fi)
__device__ __forceinline__ f4 cscale(f4 v, float fr, float fi) {
  f4 r;
  r.x = fr * v.x - fi * v.y;
  r.y = fr * v.y + fi * v.x;
  r.z = fr * v.z - fi * v.w;
  r.w = fr * v.w + fi * v.z;
  return r;
}

// control-gate factor: 1 if x-bit==0, else (+i if d-bit else -i)
__device__ __forceinline__ void fac(int xbit, int dbit, float& fr, float& fi) {
  fr = xbit ? 0.0f : 1.0f;
  fi = xbit ? (dbit ? 1.0f : -1.0f) : 0.0f;
}

__device__ __forceinline__ void ntst(float* out, unsigned idx4, f4 v) {
  __builtin_nontemporal_store(v, ((f4*)out) + idx4);
}

// Fused kernel. 1536 workgroups x 256 threads:
//   WG [   0, 512)  -> block0
//   WG [ 512,1024)  -> block1
//   WG [1024,1280)  -> block2 (async-to-LDS staged 1024-complex tile)
//   WG [1280,1536)  -> block3 (free-bit replication spread across threads)
__global__ __launch_bounds__(256) void k_entangle(const float* __restrict__ in,
                                                  float* __restrict__ out) {
  __shared__ f4 sm[512];                          // 8 KiB, used by block2 WGs
  const f4* in4 = (const f4*)in;
  unsigned wg = blockIdx.x, tid = threadIdx.x;

  if (wg < 512u) {
    // ---------------- block 0: two diagonal scales ---------------------------
    unsigned t = wg * 256u + tid;                 // 131072 threads
    unsigned b = t >> 13, r = t & 8191u;          // r = f4 index, x = 2r
    f4 v = in4[(b * 4u + 0u) * DIM4 + r];
    int bit13 = (r >> 12) & 1;                    // xbit13
    int bit6  = (r >> 5) & 1;                     // xbit6
#pragma unroll
    for (int d0 = 0; d0 < 2; ++d0)
#pragma unroll
      for (int d2 = 0; d2 < 2; ++d2) {
        float ar, ai, br, bi;
        fac(bit13, d0, ar, ai);
        fac(bit6, d2, br, bi);
        float fr = ar * br - ai * bi, fi = ar * bi + ai * br;
        f4 o = cscale(v, fr, fi);
        unsigned base = (b * 64u + (unsigned)(d0 + 4 * d2)) * DIM4 + r;
        ntst(out, base + 0u * DIM4, o);           // free d-bits 1,3 -> +{0,2,8,10}
        ntst(out, base + 2u * DIM4, o);
        ntst(out, base + 8u * DIM4, o);
        ntst(out, base + 10u * DIM4, o);
      }

  } else if (wg < 1024u) {
    // ---------------- block 1: dense on x-bit0 (inside one f4), scale bit11 --
    unsigned t = (wg - 512u) * 256u + tid;        // 131072 threads
    unsigned b = t >> 13, r = t & 8191u;
    f4 v = in4[(b * 4u + 1u) * DIM4 + r];
    f4 oI;                                        // tI applied across lanes
    oI.x = 0.5f * (v.x + v.y + v.z - v.w);
    oI.y = 0.5f * (v.y - v.x + v.z + v.w);
    oI.z = 0.5f * (v.z + v.w + v.x - v.y);
    oI.w = 0.5f * (v.w - v.z + v.x + v.y);
    f4 oZ = __builtin_shufflevector(oI, oI, 2, 3, 0, 1);  // tZ = swapped outs
    int bit11 = (r >> 10) & 1;
#pragma unroll
    for (int d1 = 0; d1 < 2; ++d1) {
      float sr, si;
      fac(bit11, d1, sr, si);
#pragma unroll
      for (int d0 = 0; d0 < 2; ++d0) {
        f4 o = cscale(d0 ? oZ : oI, sr, si);
        unsigned base = (b * 64u + 16u + (unsigned)(d0 + 2 * d1)) * DIM4 + r;
        ntst(out, base + 0u * DIM4, o);           // free d-bits 2,3 -> +{0,4,8,12}
        ntst(out, base + 4u * DIM4, o);
        ntst(out, base + 8u * DIM4, o);
        ntst(out, base + 12u * DIM4, o);
      }
    }

  } else if (wg < 1280u) {
    // ---------------- block 2: async-LDS tile, dense bit8, scale bit3 --------
    unsigned wg2 = wg - 1024u;                    // 256 WGs
    unsigned b = wg2 >> 4, tile = wg2 & 15u;      // 1024-complex tile
    const f4* gp = in4 + (b * 4u + 2u) * DIM4 + tile * 512u;
#if defined(__gfx1250__)
    // CDNA5 async DMA into LDS (ASYNCcnt-tracked, bypasses VGPRs). LDS byte
    // offset = low 32 bits of the generic pointer to __shared__.
    unsigned lds0 = (unsigned)(unsigned long long)(&sm[0]);
    {
      unsigned loff0 = lds0 + tid * 16u;
      const f4* ga0 = gp + tid;
      asm volatile("global_load_async_to_lds_b128 %0, %1, off"
                   :: "v"(loff0), "v"(ga0) : "memory");
      unsigned loff1 = lds0 + (tid + 256u) * 16u;
      const f4* ga1 = gp + tid + 256u;
      asm volatile("global_load_async_to_lds_b128 %0, %1, off"
                   :: "v"(loff1), "v"(ga1) : "memory");
      asm volatile("s_wait_asynccnt 0" ::: "memory");
    }
#else
    sm[tid] = gp[tid];
    sm[tid + 256u] = gp[tid + 256u];
#endif
    __syncthreads();
    unsigned c = tid >> 7, lo = tid & 127u;       // 512-chunk, pair index
    f4 vA = sm[c * 256u + lo];                    // positions p,p+1   (bit8=0)
    f4 vB = sm[c * 256u + lo + 128u];             // positions p+256.. (bit8=1)
    f4 O0 = cmix(vA, vB);                         // tI out at bit8=0
    f4 O1 = cmix(vB, vA);                         // tI out at bit8=1
    int bit3 = (lo >> 2) & 1;                     // bit3 of p = 2*lo
    unsigned g = tile * 512u + c * 256u + lo;     // f4 index of low pair
#pragma unroll
    for (int d3 = 0; d3 < 2; ++d3) {
      float sr, si;
      fac(bit3, d3, sr, si);
#pragma unroll
      for (int d1 = 0; d1 < 2; ++d1) {
        f4 lov = cscale(d1 ? O1 : O0, sr, si);    // tZ swaps the two outputs
        f4 hiv = cscale(d1 ? O0 : O1, sr, si);
        unsigned d = 2u * (unsigned)d1 + 8u * (unsigned)d3;
        unsigned base = (b * 64u + 32u + d) * DIM4 + g;
        ntst(out, base + 0u * DIM4, lov);         // free d-bits 0,2 -> +{0,1,4,5}
        ntst(out, base + 0u * DIM4 + 128u, hiv);
        ntst(out, base + 1u * DIM4, lov);
        ntst(out, base + 1u * DIM4 + 128u, hiv);
        ntst(out, base + 4u * DIM4, lov);
        ntst(out, base + 4u * DIM4 + 128u, hiv);
        ntst(out, base + 5u * DIM4, lov);
        ntst(out, base + 5u * DIM4 + 128u, hiv);
      }
    }

  } else {
    // ---------------- block 3: dense on x-bits 12 and 1 ----------------------
    // fb (one free-d-bit replica) in HIGH index bits: 4x parallelism, loads and
    // stores stay coalesced; redundant reads hit L2 (input slice is 2 MiB).
    unsigned t = (wg - 1280u) * 256u + tid;       // 65536 threads
    unsigned b = t >> 12;                         // 4096 threads per batch
    unsigned rem = t & 4095u;
    unsigned fb = rem >> 10;                      // free d-bits 0,1 -> d += fb
    unsigned u = rem & 1023u;                     // unit: p = 4u (bit12=0)
    unsigned q = u * 2u;                          // f4 index of p
    const f4* ip = in4 + (b * 4u + 3u) * DIM4;
    f4 L0 = ip[q], L1 = ip[q + 1u];               // p..p+3       (bit12=0)
    f4 H0 = ip[q + 2048u], H1 = ip[q + 2049u];    // p+4096..     (bit12=1)
    f4 A0 = cmix(L0, H0), A1 = cmix(L1, H1);      // tI@bit12: low-side results
    f4 B0 = cmix(H0, L0), B1 = cmix(H1, L1);      //           high-side results
#pragma unroll
    for (int d2 = 0; d2 < 2; ++d2) {
      f4 G0 = d2 ? B0 : A0, G1 = d2 ? B1 : A1;    // bit12=0 group (tZ swaps)
      f4 K0 = d2 ? A0 : B0, K1 = d2 ? A1 : B1;    // bit12=1 group
      f4 M0 = cmix(G0, G1), M1 = cmix(G1, G0);    // tI@bit1 on low group
      f4 N0 = cmix(K0, K1), N1 = cmix(K1, K0);    // tI@bit1 on high group
#pragma unroll
      for (int d3 = 0; d3 < 2; ++d3) {
        f4 lo01 = d3 ? M1 : M0, lo23 = d3 ? M0 : M1;
        f4 hi01 = d3 ? N1 : N0, hi23 = d3 ? N0 : N1;
        unsigned d = 4u * (unsigned)d2 + 8u * (unsigned)d3 + fb;
        unsigned o = (b * 64u + 48u + d) * DIM4 + q;
        ntst(out, o, lo01);
        ntst(out, o + 1u, lo23);
        ntst(out, o + 2048u, hi01);
        ntst(out, o + 2049u, hi23);
      }
    }
  }
}

extern "C" void kernel_launch(void* const* d_in, const int* in_sizes, int n_in,
                              void* d_out, int out_size, void* d_ws, size_t ws_size,
                              hipStream_t stream) {
  (void)in_sizes; (void)n_in; (void)out_size; (void)d_ws; (void)ws_size;
  const float* in = (const float*)d_in[0];   // complex64 -> interleaved f32
  float* out = (float*)d_out;                // (16,4,16,16384) complex64
  k_entangle<<<1536, 256, 0, stream>>>(in, out);
}